// MultiSubjectMultiTexture_32177894982386
// MI455X (gfx1250) — compile-verified
//
#include <hip/hip_runtime.h>

// Problem constants (match reference)
#define N_  4
#define H_  512
#define W_  512
#define C_  8
#define HW_ (H_ * W_)   // 262144 = 2^18

// 8-byte texel pair with only 4-byte alignment guarantee (xb may be odd).
struct __align__(4) F2 { float a, b; };

__global__ __launch_bounds__(256) void multisubject_grid_sample_kernel(
    const float* __restrict__ uv,     // [N,H,W,2]
    const int*   __restrict__ idcs,   // [N,H,W]
    const int*   __restrict__ sid,    // [N]
    const float* __restrict__ tex0,   // [M,C,512,512]
    const float* __restrict__ tex1,   // [M,C,512,512]
    float*       __restrict__ out)    // [N,C,H,W]
{
    __shared__ alignas(16) unsigned char smem[256 * 8 + 256 * 4];

    const unsigned tid = threadIdx.x;
    const unsigned i   = blockIdx.x * 256u + tid;   // global pixel id

    // ---- CDNA5 async stage-in of streaming inputs (ASYNCcnt path) ----
    // Each lane pulls its own uv (8B) and idx (4B) straight into LDS via the
    // async engine, bypassing the VGPR return path.
    const unsigned lds_uv = (unsigned)(size_t)(&smem[tid * 8]);
    const unsigned lds_id = (unsigned)(size_t)(&smem[2048 + tid * 4]);
    const float* guv = uv + (size_t)i * 2;
    const int*   gix = idcs + i;
    asm volatile(
        "global_load_async_to_lds_b64 %0, %1, off\n\t"
        "global_load_async_to_lds_b32 %2, %3, off"
        :
        : "v"(lds_uv), "v"(guv), "v"(lds_id), "v"(gix)
        : "memory");
    asm volatile("s_wait_asynccnt 0x0" ::: "memory");
    // Each lane reads back only its own slots -> no cross-wave barrier needed.

    const float u = ((const float*)smem)[tid * 2 + 0];
    const float v = ((const float*)smem)[tid * 2 + 1];
    const int idx = *(const int*)(&smem[2048 + tid * 4]);

    // align_corners=True mapping with border clamp (matches reference exactly)
    float x = (u + 1.0f) * 0.5f * (float)(W_ - 1);
    float y = (v + 1.0f) * 0.5f * (float)(H_ - 1);
    x = fminf(fmaxf(x, 0.0f), (float)(W_ - 1));
    y = fminf(fmaxf(y, 0.0f), (float)(H_ - 1));
    const float xf = floorf(x);
    const float yf = floorf(y);
    const int x0 = (int)xf;
    const int y0 = (int)yf;
    const int y1 = (y0 < H_ - 1) ? y0 + 1 : H_ - 1;
    const int xb = (x0 < W_ - 1) ? x0 : W_ - 2;   // clamped pair base
    const bool xe = (x0 == W_ - 1);               // x-border: both taps = pair.b
    const float wx = x - xf;
    const float wy = y - yf;
    const float cwx = 1.0f - wx;
    const float cwy = 1.0f - wy;

    const unsigned n = i >> 18;            // HW_ = 2^18  (block-uniform)
    const unsigned p = i & (HW_ - 1);      // h*W + w
    const int s = sid[n];

    // Sample only the selected bank (reference samples both, then selects).
    const float* tb = (idx ? tex1 : tex0) + (size_t)s * (size_t)(C_ * HW_);
    const float* r0 = tb + (unsigned)(y0 * W_ + xb);   // row y0, cols xb..xb+1
    const float* r1 = tb + (unsigned)(y1 * W_ + xb);   // row y1
    float* ob = out + (size_t)n * (size_t)(C_ * HW_) + p;

    // 8 channels: 16 b64 gathers off 2 address registers, channel stride
    // c*HW_*4 = c MiB folds into the signed 24-bit instruction offset.
#pragma unroll
    for (int c = 0; c < C_; ++c) {
        const F2 t = *(const F2*)(r0 + c * HW_);
        const F2 b = *(const F2*)(r1 + c * HW_);
        const float v00 = xe ? t.b : t.a;
        const float v10 = xe ? b.b : b.a;
        const float top = v00 * cwx + t.b * wx;
        const float bot = v10 * cwx + b.b * wx;
        // Output is write-once (32 MB): NT store keeps L2 for the hot textures.
        __builtin_nontemporal_store(top * cwy + bot * wy, ob + c * HW_);
    }
}

extern "C" void kernel_launch(void* const* d_in, const int* in_sizes, int n_in,
                              void* d_out, int out_size, void* d_ws, size_t ws_size,
                              hipStream_t stream) {
    (void)in_sizes; (void)n_in; (void)out_size; (void)d_ws; (void)ws_size;
    const float* uv   = (const float*)d_in[0];
    const int*   idcs = (const int*)d_in[1];
    const int*   sid  = (const int*)d_in[2];
    const float* tex0 = (const float*)d_in[3];
    const float* tex1 = (const float*)d_in[4];
    float* out = (float*)d_out;

    const int total_pixels = N_ * HW_;            // 1,048,576
    const int blocks = total_pixels / 256;        // 4096 blocks x 8 wave32
    multisubject_grid_sample_kernel<<<blocks, 256, 0, stream>>>(
        uv, idcs, sid, tex0, tex1, out);
}